// CharDistributionAnalyzer_23957327577718
// MI455X (gfx1250) — compile-verified
//
#include <hip/hip_runtime.h>

// CharDistributionAnalyzer for MI455X (gfx1250).
// Memory-bound streaming histogram: 128 MiB in / 6 MiB out -> ~6us floor at
// 23.3 TB/s. Strategy: async global->LDS b128 copies (coalesced, tracked by
// ASYNCcnt), LDS ds_add_u32 histogram with bank-padded stride, per-row scalar
// feature pass. No matmul structure exists here, so no WMMA by design.

#define SEQ_LEN        128
#define VOCAB          40
#define ROWS_PER_BLOCK 32
#define NTHREADS       256
#define HSTRIDE        48   // padded histogram row stride in dwords (breaks bank alignment)
#define TOK_BYTES      (ROWS_PER_BLOCK * SEQ_LEN * 4)   // 16384 bytes per block tile

// The builtin's diagnostic revealed its parameter types:
//   arg0: 'int __vector_size__(16)' pointer in global (__device__) AS
//   arg1: same vector type in LDS AS
typedef int v4i __attribute__((vector_size(16)));
typedef __attribute__((address_space(1))) v4i as1_v4i;
typedef __attribute__((address_space(3))) v4i as3_v4i;
typedef __attribute__((address_space(3))) void as3_void;

__device__ __forceinline__ void async_load_b128_to_lds(const void* gp, void* lp) {
#if __has_builtin(__builtin_amdgcn_global_load_async_to_lds_b128)
  __builtin_amdgcn_global_load_async_to_lds_b128(
      (as1_v4i*)gp, (as3_v4i*)lp, 0, 0);
#else
  unsigned           loff = (unsigned)(size_t)(as3_void*)lp;
  unsigned long long ga   = (unsigned long long)gp;
  asm volatile("global_load_async_to_lds_b128 %0, %1, off"
               :: "v"(loff), "v"(ga)
               : "memory");
#endif
}

__device__ __forceinline__ void wait_asynccnt_zero() {
#if __has_builtin(__builtin_amdgcn_s_wait_asynccnt)
  __builtin_amdgcn_s_wait_asynccnt(0);
#else
  asm volatile("s_wait_asynccnt 0" ::: "memory");
#endif
}

__global__ __launch_bounds__(NTHREADS)
void char_dist_feats_kernel(const int* __restrict__ x,
                            float* __restrict__ out,
                            int nrows) {
  __shared__ __align__(16) int      toks[ROWS_PER_BLOCK * SEQ_LEN];   // 16 KB raw token tile
  __shared__ __align__(16) unsigned hist[ROWS_PER_BLOCK * HSTRIDE];   // 6 KB padded histograms

  const int tid  = threadIdx.x;
  const int row0 = blockIdx.x * ROWS_PER_BLOCK;

  // ---------- Phase 0: async copy of 32 contiguous rows (16 KB) global->LDS.
  // Consecutive tids hit consecutive 16B chunks: perfectly coalesced b128s.
  {
    const char* gbase = (const char*)x + (size_t)row0 * (SEQ_LEN * 4);
    char*       lbase = (char*)toks;
#pragma unroll
    for (int k = 0; k < TOK_BYTES / (NTHREADS * 16); ++k) {   // 4 iterations
      const int off = (k * NTHREADS + tid) * 16;
      async_load_b128_to_lds(gbase + off, lbase + off);
    }
  }

  // Clear histograms while the async copies are in flight (disjoint LDS region).
#pragma unroll
  for (int k = 0; k < (ROWS_PER_BLOCK * HSTRIDE) / NTHREADS; ++k)   // 6 iterations
    hist[k * NTHREADS + tid] = 0u;

  wait_asynccnt_zero();   // this wave's async tile data is in LDS
  __syncthreads();        // all waves' data + cleared hist visible block-wide

  // ---------- Phase 1: histogram accumulate. 8 threads per row, 16 tokens each.
  {
    const int r = tid >> 3;        // row in tile: 0..31
    const int p = tid & 7;         // part within row: 0..7
    if (row0 + r < nrows) {
      const uint4* tp = (const uint4*)&toks[r * SEQ_LEN + p * 16];
      unsigned*    h  = &hist[r * HSTRIDE];
#pragma unroll
      for (int k = 0; k < 4; ++k) {
        const uint4 t = tp[k];
        // token in [1, VOCAB) counts; 0 is padding and bin 0 stays 0.
        if (t.x - 1u < (unsigned)(VOCAB - 1)) atomicAdd(&h[t.x], 1u);
        if (t.y - 1u < (unsigned)(VOCAB - 1)) atomicAdd(&h[t.y], 1u);
        if (t.z - 1u < (unsigned)(VOCAB - 1)) atomicAdd(&h[t.z], 1u);
        if (t.w - 1u < (unsigned)(VOCAB - 1)) atomicAdd(&h[t.w], 1u);
      }
    }
  }
  __syncthreads();   // drains DScnt (atomics) before the feature pass reads

  // ---------- Phase 2: per-row feature extraction, one lane per row.
  if (tid < ROWS_PER_BLOCK) {
    const int row = row0 + tid;
    const uint4* hv = (const uint4*)&hist[tid * HSTRIDE];

    unsigned c[VOCAB];
#pragma unroll
    for (int k = 0; k < VOCAB / 4; ++k) {    // 10x ds_load_b128
      const uint4 u = hv[k];
      c[4 * k + 0] = u.x;
      c[4 * k + 1] = u.y;
      c[4 * k + 2] = u.z;
      c[4 * k + 3] = u.w;
    }

    int total = 0, uniq = 0, cmax = 0, cmin = 0x7fffffff;
    int letters = 0, digits = 0, special = 0;
#pragma unroll
    for (int v = 1; v < VOCAB; ++v) {
      const int cc = (int)c[v];
      total += cc;
      uniq  += (cc > 0) ? 1 : 0;
      cmax   = (cc > cmax) ? cc : cmax;
      cmin   = (cc > 0 && cc < cmin) ? cc : cmin;
      if (v < 27)      letters += cc;   // bins 1..26
      else if (v < 37) digits  += cc;   // bins 27..36
      else             special += cc;   // bins 37..39
    }

    float f0, f1, f2, f3, f4, f5;
    if (total > 0) {
      const float ft = (float)total;
      f0 = (float)uniq    / (float)VOCAB;
      f1 = (float)cmax    / ft;
      f2 = (float)cmin    / ft;
      f3 = (float)letters / ft;
      f4 = (float)digits  / ft;
      f5 = (float)special / ft;
    } else {
      f0 = f1 = f2 = f3 = f4 = f5 = 0.0f;
    }

    if (row < nrows) {
      float* o = out + (size_t)row * 6;
      o[0] = f0; o[1] = f1; o[2] = f2;
      o[3] = f3; o[4] = f4; o[5] = f5;
    }
  }
}

extern "C" void kernel_launch(void* const* d_in, const int* in_sizes, int n_in,
                              void* d_out, int out_size, void* d_ws, size_t ws_size,
                              hipStream_t stream) {
  (void)n_in; (void)out_size; (void)d_ws; (void)ws_size;
  const int* x   = (const int*)d_in[0];
  float*     out = (float*)d_out;
  const int  nrows = in_sizes[0] / SEQ_LEN;          // 262144 for the reference shapes
  const int  grid  = (nrows + ROWS_PER_BLOCK - 1) / ROWS_PER_BLOCK;  // 8192 (divides exactly)
  char_dist_feats_kernel<<<grid, NTHREADS, 0, stream>>>(x, out, nrows);
}